// CTCLoss_28707561407049
// MI455X (gfx1250) — compile-verified
//
#include <hip/hip_runtime.h>
#include <stdint.h>

#define T_DIM 256
#define B_DIM 64
#define C_DIM 6625
#define S_DIM 32
#define K_DIM 65          // 2*S+1
#define NEGV  (-1e30f)

// ---------------------------------------------------------------------------
// CDNA5 async-to-LDS helpers (ASYNCcnt-tracked DMA; CDNA5 ISA)
// ---------------------------------------------------------------------------
__device__ __forceinline__ void async_b128(unsigned lds_addr, uint64_t gaddr) {
  // vdst = LDS byte address, vaddr = 64-bit global address (GV mode)
  asm volatile("global_load_async_to_lds_b128 %0, %1, off"
               :: "v"(lds_addr), "v"(gaddr)
               : "memory");
}
__device__ __forceinline__ void wait_async_le4() {
  asm volatile("s_wait_asynccnt 4" ::: "memory");
}
__device__ __forceinline__ void wait_async_le0() {
  asm volatile("s_wait_asynccnt 0" ::: "memory");
}

// branch-free 2-exp combine (used only for the few lead/tail elements)
__device__ __forceinline__ void acc2(float v, float& m, float& s) {
  const float M = fmaxf(m, v);
  s = s * __expf(m - M) + __expf(v - M);
  m = M;
}

__device__ __forceinline__ float lae(float a, float b) {  // logaddexp
  const float M  = fmaxf(a, b);
  const float mn = fminf(a, b);
  return M + __logf(1.0f + __expf(mn - M));
}

// ---------------------------------------------------------------------------
// Kernel 1: lse[t*B+b] = logsumexp_c preds[t,b,:]   (one wave per row)
// Streams 434 MB of preds through LDS with double-buffered async b128 DMA.
// Hot loop is branch-free: per-16-element max (VALU) then ~1.06 exp/element.
// ---------------------------------------------------------------------------
#define WAVES_PER_BLK 8
#define TILE_FLOATS   512           // per-wave tile: 4 x b128 per lane = 2 KB

__global__ __launch_bounds__(256) void lse_kernel(const float* __restrict__ preds,
                                                  float* __restrict__ lse_out) {
  __shared__ float smem[WAVES_PER_BLK * 2 * TILE_FLOATS];   // 32 KB
  const int lane = threadIdx.x & 31;
  const int wave = threadIdx.x >> 5;
  const int row  = blockIdx.x * WAVES_PER_BLK + wave;       // row in [0, T*B)

  const size_t row_off = (size_t)row * C_DIM;
  const float* rp = preds + row_off;

  float m = NEGV, s = 0.0f;

  // C=6625 is odd -> row base only 4B aligned; peel to 16B alignment for b128
  const int lead = (int)((4u - ((unsigned)row_off & 3u)) & 3u);
  if (lane < lead) acc2(rp[lane], m, s);
  const float* ap = rp + lead;                               // 16B-aligned
  const int n      = C_DIM - lead;
  const int ntiles = n / TILE_FLOATS;                        // = 12
  const int tail0  = lead + ntiles * TILE_FLOATS;

  const unsigned lbase = ((unsigned)(uintptr_t)(&smem[0]))
                       + (unsigned)wave * (2u * TILE_FLOATS * 4u);
  const uint64_t gbase = (uint64_t)(uintptr_t)ap;

  auto issue_tile = [&](int t) {
    const unsigned pb = ((unsigned)t & 1u) * (TILE_FLOATS * 4u);
    const uint64_t g  = gbase + (uint64_t)t * (TILE_FLOATS * 4u);
#pragma unroll
    for (int j = 0; j < 4; ++j) {
      const unsigned off = (unsigned)(j * 512 + lane * 16);  // bytes
      async_b128(lbase + pb + off, g + off);
    }
  };

  if (ntiles > 0) issue_tile(0);
  for (int t = 0; t < ntiles; ++t) {
    if (t + 1 < ntiles) { issue_tile(t + 1); wait_async_le4(); }  // oldest 4 ops done
    else                { wait_async_le0(); }
    const float* buf = &smem[wave * (2 * TILE_FLOATS) + (t & 1) * TILE_FLOATS];

    float4 v[4];
#pragma unroll
    for (int j = 0; j < 4; ++j)
      v[j] = *(const float4*)(buf + j * 128 + lane * 4);

    // tile max (pure VALU, no divergence)
    float tm = v[0].x;
#pragma unroll
    for (int j = 0; j < 4; ++j) {
      tm = fmaxf(tm, fmaxf(fmaxf(v[j].x, v[j].y), fmaxf(v[j].z, v[j].w)));
    }
    const float M = fmaxf(m, tm);
    s *= __expf(m - M);          // one rescale per 16 elements
    m = M;
#pragma unroll
    for (int j = 0; j < 4; ++j) {
      s += __expf(v[j].x - M) + __expf(v[j].y - M)
         + __expf(v[j].z - M) + __expf(v[j].w - M);
    }
  }
  for (int i = tail0 + lane; i < C_DIM; i += 32) acc2(rp[i], m, s);

  // wave32 reduction of (m, s) pairs
#pragma unroll
  for (int off = 16; off > 0; off >>= 1) {
    const float mo = __shfl_xor(m, off, 32);
    const float so = __shfl_xor(s, off, 32);
    const float M  = fmaxf(m, mo);
    s = s * __expf(m - M) + so * __expf(mo - M);
    m = M;
  }
  if (lane == 0) lse_out[row] = m + __logf(s);
}

// ---------------------------------------------------------------------------
// Kernel 2: per-batch CTC alpha recursion (one 128-thread block per batch).
// Stage all needed logits into LDS first so the 256-step chain never waits on HBM.
// ---------------------------------------------------------------------------
__global__ __launch_bounds__(128) void ctc_kernel(const float* __restrict__ preds,
                                                  const int* __restrict__ targets,
                                                  const int* __restrict__ in_len,
                                                  const int* __restrict__ tgt_len,
                                                  const float* __restrict__ lse,
                                                  float* __restrict__ loss_out) {
  const int b   = blockIdx.x;
  const int tid = threadIdx.x;

  __shared__ int   ext[K_DIM];                 // extended label sequence
  __shared__ float lrow[T_DIM];                // lse for this batch
  __shared__ float lpb[T_DIM];                 // log-prob of blank per t (lse-subtracted)
  __shared__ float lpl[T_DIM * S_DIM];         // log-prob of label j per t (32 KB)
  __shared__ float A[2][K_DIM + 2];            // alpha double buffer, 2-elt NEG pad

  if (tid < K_DIM)
    ext[tid] = (tid & 1) ? targets[b * S_DIM + (tid >> 1)] : 0;
  for (int t = tid; t < T_DIM; t += 128) lrow[t] = lse[t * B_DIM + b];
  __syncthreads();

  const bool myskip = (tid >= 2) && (tid < K_DIM) &&
                      (ext[tid] != 0) && (ext[tid] != ext[tid - 2]);

  const size_t BC = (size_t)B_DIM * C_DIM;
  for (int t = tid; t < T_DIM; t += 128)
    lpb[t] = preds[(size_t)t * BC + (size_t)b * C_DIM + 0] - lrow[t];
#pragma unroll 8
  for (int idx = tid; idx < T_DIM * S_DIM; idx += 128) {
    const int t = idx >> 5, j = idx & (S_DIM - 1);
    lpl[idx] = preds[(size_t)t * BC + (size_t)b * C_DIM + ext[2 * j + 1]] - lrow[t];
  }
  if (tid < K_DIM + 2) { A[0][tid] = NEGV; A[1][tid] = NEGV; }
  __syncthreads();

  if (tid == 0) A[0][2] = lpb[0];              // alpha0[0] = lp(0, blank)
  if (tid == 1) A[0][3] = lpl[0];              // alpha0[1] = lp(0, y0)
  __syncthreads();

  const int ilen = in_len[b];
  int p = 0;
  for (int t = 1; t < T_DIM; ++t) {
    float na;
    if (tid < K_DIM) {
      const float a  = A[p][tid + 2];
      const float a1 = A[p][tid + 1];
      const float a2 = myskip ? A[p][tid] : NEGV;
      const float lpv = (tid & 1) ? lpl[t * S_DIM + (tid >> 1)] : lpb[t];
      const float nv = lae(lae(a, a1), a2) + lpv;
      na = (t < ilen) ? nv : a;                // freeze past input length
    }
    if (tid < K_DIM) A[1 - p][tid + 2] = na;
    __syncthreads();
    p = 1 - p;
  }

  if (tid == 0) {
    const int L = tgt_len[b];
    const float a1 = A[p][2 * L - 1 + 2];
    const float a2 = A[p][2 * L + 2];
    float loss = -lae(a1, a2);
    if (loss > 1e29f) loss = 0.0f;             // zero_infinity
    const float pt = __expf(-loss);
    const float om = 1.0f - pt;
    loss = om * om * loss;                     // focal, GAMMA = 2
    loss_out[b] = loss / (float)L / (float)B_DIM;
  }
}

// ---------------------------------------------------------------------------
// Kernel 3: sum 64 per-batch contributions -> scalar
// ---------------------------------------------------------------------------
__global__ void reduce_kernel(const float* __restrict__ loss_in, float* __restrict__ out) {
  __shared__ float sh[B_DIM];
  const int tid = threadIdx.x;
  sh[tid] = loss_in[tid];
  __syncthreads();
  for (int off = B_DIM / 2; off > 0; off >>= 1) {
    if (tid < off) sh[tid] += sh[tid + off];
    __syncthreads();
  }
  if (tid == 0) out[0] = sh[0];
}

extern "C" void kernel_launch(void* const* d_in, const int* in_sizes, int n_in,
                              void* d_out, int out_size, void* d_ws, size_t ws_size,
                              hipStream_t stream) {
  (void)in_sizes; (void)n_in; (void)out_size; (void)ws_size;
  const float* preds   = (const float*)d_in[0];
  const int*   targets = (const int*)d_in[1];
  const int*   in_len  = (const int*)d_in[2];
  const int*   tgt_len = (const int*)d_in[3];

  float* lse   = (float*)d_ws;                 // T*B floats
  float* lossb = lse + (size_t)T_DIM * B_DIM;  // B floats

  lse_kernel<<<(T_DIM * B_DIM) / WAVES_PER_BLK, 256, 0, stream>>>(preds, lse);
  ctc_kernel<<<B_DIM, 128, 0, stream>>>(preds, targets, in_len, tgt_len, lse, lossb);
  reduce_kernel<<<1, B_DIM, 0, stream>>>(lossb, (float*)d_out);
}